// Coords2Stress_9526237462644
// MI455X (gfx1250) — compile-verified
//
#include <hip/hip_runtime.h>
#include <hip/hip_bf16.h>

// Problem constants (match reference)
#define BN      2
#define NATOMS  768
#define TN      (3 * NATOMS)          // 2304
#define BOXD    80
#define BOX3    (BOXD * BOXD * BOXD)
#define CUT     15.0f
#define NV      16                    // subspace width (col 6 is the answer)
#define NITER   160                   // subspace-iteration sweeps
#define SIGMA   4096.0f               // Gershgorin-style spectral shift
#define KCHUNK  128                   // K panel staged in LDS per workgroup
#define WPB     4                     // waves per gemm workgroup
#define APAD    4                     // LDS row padding (dwords) -> no bank conflicts

// gfx1250 data-mover switches (flip to 0 if the assembler rejects a mnemonic)
#define GEMM_ASYNC_LDS 1              // global_load_async_to_lds_b128 (ASYNCcnt)
#define GEMM_TDM       1              // tensor_load_to_lds (TENSORcnt)

typedef __attribute__((ext_vector_type(2))) float v2f;
typedef __attribute__((ext_vector_type(8))) float v8f;
typedef __attribute__((ext_vector_type(4))) unsigned int u32x4;
typedef __attribute__((ext_vector_type(8))) unsigned int u32x8;

// ---------------------------------------------------------------------------
// 1) Hessian build: one wave per (batch, atom i). 42.5 MB of stores -> BW bound.
// ---------------------------------------------------------------------------
__global__ void __launch_bounds__(128)
hess_kernel(const float* __restrict__ coords, float* __restrict__ H) {
  int gw   = (blockIdx.x * blockDim.x + threadIdx.x) >> 5;
  int lane = threadIdx.x & 31;
  if (gw >= BN * NATOMS) return;
  int b = gw / NATOMS, i = gw % NATOMS;
  const float* c = coords + (size_t)b * (3 * NATOMS);
  float xi = c[3 * i + 0], yi = c[3 * i + 1], zi = c[3 * i + 2];
  float* Hb = H + (size_t)b * TN * TN;
  float dg[9] = {0, 0, 0, 0, 0, 0, 0, 0, 0};
  for (int j0 = 0; j0 < NATOMS; j0 += 32) {
    int j = j0 + lane;
    float sx = xi - c[3 * j + 0];
    float sy = yi - c[3 * j + 1];
    float sz = zi - c[3 * j + 2];
    float d2 = sx * sx + sy * sy + sz * sz + 1e-5f;
    float w  = (sqrtf(d2) < CUT) ? (-1.0f / d2) : 0.0f;
    float h[9] = { w * sx * sx, w * sx * sy, w * sx * sz,
                   w * sy * sx, w * sy * sy, w * sy * sz,
                   w * sz * sx, w * sz * sy, w * sz * sz };
#pragma unroll
    for (int k = 0; k < 9; ++k) dg[k] -= h[k];
    if (j != i) {
#pragma unroll
      for (int a = 0; a < 3; ++a)
#pragma unroll
        for (int cc = 0; cc < 3; ++cc)
          Hb[(size_t)(3 * i + a) * TN + (3 * j + cc)] = h[a * 3 + cc];
    }
  }
#pragma unroll
  for (int k = 0; k < 9; ++k)
#pragma unroll
    for (int off = 16; off > 0; off >>= 1)
      dg[k] += __shfl_xor(dg[k], off, 32);
  if (lane == 0) {
#pragma unroll
    for (int a = 0; a < 3; ++a)
#pragma unroll
      for (int cc = 0; cc < 3; ++cc)
        Hb[(size_t)(3 * i + a) * TN + (3 * i + cc)] = dg[a * 3 + cc];
  }
}

// ---------------------------------------------------------------------------
// 2) Deterministic pseudo-random subspace init
// ---------------------------------------------------------------------------
__global__ void init_x_kernel(float* __restrict__ X) {
  int i = blockIdx.x * blockDim.x + threadIdx.x;
  if (i >= BN * TN * NV) return;
  unsigned u = (unsigned)i * 2654435761u + 12345u;
  u ^= u >> 13; u *= 2246822519u; u ^= u >> 16;
  X[i] = (float)(u & 0xFFFFFF) * (1.0f / 8388608.0f) - 1.0f;
}

// ---------------------------------------------------------------------------
// 3) Y = SIGMA*X - H*X via V_WMMA_F32_16X16X4_F32.
// Per chunk: each wave DMAs its strided 16xKCHUNK A-tile of H into LDS with
// tensor_load_to_lds (TDM, TENSORcnt; D# pads rows by 4 dwords so ds reads are
// bank-conflict-free), the block stages the shared X panel with
// global_load_async_to_lds_b128 (ASYNCcnt); the K loop is then pure
// ds_load + wmma. Sigma shift applied in the epilogue.
// ---------------------------------------------------------------------------
__global__ void __launch_bounds__(WPB * 32)
gemm_kernel(const float* __restrict__ H, const float* __restrict__ X,
            float* __restrict__ Y) {
  __shared__ float As[WPB][16 * (KCHUNK + APAD)];   // 33.8 KB A tiles
  __shared__ float Xs[KCHUNK * NV];                 // 8 KB X panel
  const int tid  = threadIdx.x;
  const int lane = tid & 31;
  const int wid  = tid >> 5;                        // 0..WPB-1
  const int row0 = (blockIdx.x * WPB + wid) * 16;
  const int b    = blockIdx.y;
  const int m    = lane & 15;                       // M row (A) / N col (B,D)
  const int kh   = (lane >> 4) << 1;                // K sub-offset: 0 or 2
  const float* Hb = H + (size_t)b * TN * TN;
  const float* Xb = X + (size_t)b * TN * NV;
  float*       Yb = Y + (size_t)b * TN * NV;
  const float* Aw = &As[wid][(size_t)m * (KCHUNK + APAD)];
  v8f acc = {};
  for (int kc = 0; kc < TN; kc += KCHUNK) {
#if GEMM_TDM
    // ---- TDM: DMA this wave's 16 x KCHUNK strided tile of H into LDS ----
    {
      unsigned long long ga =
          (unsigned long long)(const void*)(Hb + (size_t)row0 * TN + kc);
      u32x4 g0;
      g0[0] = 1u;                                   // count=1 (valid), user mode
      g0[1] = (unsigned)(size_t)&As[wid][0];        // lds_addr
      g0[2] = (unsigned)ga;                         // global_addr[31:0]
      g0[3] = (unsigned)((ga >> 32) & 0x1ffffffu)   // global_addr[56:32]
            | (2u << 30);                           // type = 2 ("image")
      u32x8 g1;
      g1[0] = (2u << 16)                            // data_size = 4 bytes
            | (1u << 20)                            // pad_enable
            | (6u << 22)                            // pad_interval: 128 dwords
            | (3u << 25);                           // pad_amount: 4 dwords
      g1[1] = ((unsigned)TN & 0xffffu) << 16;       // tensor_dim0[15:0]
      g1[2] = (((unsigned)TN >> 16) & 0xffffu)      // tensor_dim0[31:16]
            | (((unsigned)TN & 0xffffu) << 16);     // tensor_dim1[15:0]
      g1[3] = (((unsigned)TN >> 16) & 0xffffu)      // tensor_dim1[31:16]
            | ((unsigned)KCHUNK << 16);             // tile_dim0 = KCHUNK
      g1[4] = 16u;                                  // tile_dim1 = 16 rows
      g1[5] = (unsigned)TN;                         // tensor_dim0_stride[31:0]
      g1[6] = 0u;
      g1[7] = 0u;
      asm volatile("tensor_load_to_lds %0, %1" :: "s"(g0), "s"(g1) : "memory");
    }
#endif
    // ---- stage shared X[kc .. kc+KCHUNK) x NV panel into LDS ------------
    {
      const float* gsrc = Xb + (size_t)kc * NV;
#pragma unroll
      for (int v = 0; v < (KCHUNK * NV) / (WPB * 32 * 4); ++v) {
        int idx4 = (v * (WPB * 32) + tid) * 4;      // float index, 16B granules
#if GEMM_ASYNC_LDS
        unsigned lds_off = (unsigned)(size_t)&Xs[idx4];
        const float* g = gsrc + idx4;
        asm volatile("global_load_async_to_lds_b128 %0, %1, off"
                     :: "v"(lds_off), "v"(g) : "memory");
#else
        float4 d = *(const float4*)(gsrc + idx4);
        *(float4*)&Xs[idx4] = d;
#endif
      }
#if GEMM_ASYNC_LDS
      asm volatile("s_wait_asynccnt 0x0" ::: "memory");
#endif
    }
#if GEMM_TDM
    __builtin_amdgcn_s_wait_tensorcnt(0);
#endif
    __syncthreads();
#if !GEMM_TDM
    __builtin_prefetch((const void*)(Hb + (size_t)(row0 + m) * TN + kc + KCHUNK), 0, 1);
#endif
    // ---- K loop: pure ds_load + wmma ------------------------------------
#pragma unroll 4
    for (int k0 = 0; k0 < KCHUNK; k0 += 4) {
      const int k = k0 + kh;
      v2f a;                                        // A: H[row0+m][kc+k .. +1]
#if GEMM_TDM
      a.x = Aw[k];
      a.y = Aw[k + 1];
#else
      const float* Hrow = Hb + (size_t)(row0 + m) * TN;
      a.x = Hrow[kc + k];
      a.y = Hrow[kc + k + 1];
#endif
      v2f bb;                                       // B: X[kc+k][m], X[kc+k+1][m]
      bb.x = Xs[k * NV + m];
      bb.y = Xs[(k + 1) * NV + m];
      acc = __builtin_amdgcn_wmma_f32_16x16x4_f32(
          false, a, false, bb, (short)0, acc, false, false);
    }
    __syncthreads();
  }
  // D layout: lanes 0-15 -> M=r, N=lane ; lanes 16-31 -> M=8+r, N=lane-16
  const int rbase = (lane >> 4) << 3;
#pragma unroll
  for (int r = 0; r < 8; ++r) {
    const int row = row0 + rbase + r;
    Yb[(size_t)row * NV + m] = SIGMA * Xb[(size_t)row * NV + m] - acc[r];
  }
}

// ---------------------------------------------------------------------------
// 4) Modified Gram-Schmidt over the NV columns: Xout = orth(Yin).
// ---------------------------------------------------------------------------
__global__ void __launch_bounds__(256)
ortho_kernel(const float* __restrict__ Yin, float* __restrict__ Xout) {
  __shared__ float red[256];
  const int b = blockIdx.x, t = threadIdx.x;
  const float* Yb = Yin  + (size_t)b * TN * NV;
  float*       Xb = Xout + (size_t)b * TN * NV;
  if (Yb != (const float*)Xb) {
    for (int idx = t; idx < TN * NV; idx += 256) Xb[idx] = Yb[idx];
  }
  __syncthreads();
  for (int j = 0; j < NV; ++j) {
    for (int i = 0; i < j; ++i) {
      float p = 0.f;
      for (int k = t; k < TN; k += 256)
        p += Xb[(size_t)k * NV + i] * Xb[(size_t)k * NV + j];
      red[t] = p; __syncthreads();
      for (int s = 128; s > 0; s >>= 1) { if (t < s) red[t] += red[t + s]; __syncthreads(); }
      float r = red[0]; __syncthreads();
      for (int k = t; k < TN; k += 256)
        Xb[(size_t)k * NV + j] -= r * Xb[(size_t)k * NV + i];
      __syncthreads();
    }
    float p = 0.f;
    for (int k = t; k < TN; k += 256) { float v = Xb[(size_t)k * NV + j]; p += v * v; }
    red[t] = p; __syncthreads();
    for (int s = 128; s > 0; s >>= 1) { if (t < s) red[t] += red[t + s]; __syncthreads(); }
    float inv = rsqrtf(fmaxf(red[0], 1e-20f)); __syncthreads();
    for (int k = t; k < TN; k += 256) Xb[(size_t)k * NV + j] *= inv;
    __syncthreads();
  }
}

// ---------------------------------------------------------------------------
// 5) Finalize: disp = column 6 of X ; lambda = x6^T H x6 (Rayleigh quotient)
// ---------------------------------------------------------------------------
__global__ void __launch_bounds__(256)
finalize_kernel(const float* __restrict__ H, const float* __restrict__ X,
                float* __restrict__ disp, float* __restrict__ lam) {
  __shared__ float red[256];
  const int b = blockIdx.x, t = threadIdx.x;
  const float* Hb = H + (size_t)b * TN * TN;
  const float* Xb = X + (size_t)b * TN * NV;
  float part = 0.f;
  for (int r = t; r < TN; r += 256) {
    float xr = Xb[(size_t)r * NV + 6];
    disp[(size_t)b * TN + r] = xr;
    float s = 0.f;
    for (int j = 0; j < TN; ++j)
      s += Hb[(size_t)r * TN + j] * Xb[(size_t)j * NV + 6];
    part += xr * s;
  }
  red[t] = part; __syncthreads();
  for (int s = 128; s > 0; s >>= 1) { if (t < s) red[t] += red[t + s]; __syncthreads(); }
  if (t == 0) lam[b] = red[0];
}

// ---------------------------------------------------------------------------
// 6) Rasterize: zero the volume region then scatter with f32 global atomics
// ---------------------------------------------------------------------------
__global__ void zero_vol_kernel(float* __restrict__ vol) {
  size_t i = (size_t)blockIdx.x * blockDim.x + threadIdx.x;
  if (i < (size_t)BN * 3 * BOX3) vol[i] = 0.f;
}

__global__ void raster_kernel(const float* __restrict__ coords,
                              const float* __restrict__ disp,
                              float* __restrict__ vol) {
  int g = blockIdx.x * blockDim.x + threadIdx.x;
  if (g >= BN * NATOMS) return;
  int b = g / NATOMS, i = g % NATOMS;
  const float* c = coords + (size_t)b * 3 * NATOMS + 3 * i;
  int ix = min(max((int)floorf(c[0]), 0), BOXD - 1);
  int iy = min(max((int)floorf(c[1]), 0), BOXD - 1);
  int iz = min(max((int)floorf(c[2]), 0), BOXD - 1);
  size_t cell = ((size_t)ix * BOXD + iy) * BOXD + iz;
  size_t base = (size_t)b * 3 * BOX3;
#pragma unroll
  for (int ch = 0; ch < 3; ++ch)
    atomicAdd(&vol[base + (size_t)ch * BOX3 + cell],
              disp[(size_t)b * TN + 3 * i + ch] * 50.0f);
}

// ---------------------------------------------------------------------------
extern "C" void kernel_launch(void* const* d_in, const int* in_sizes, int n_in,
                              void* d_out, int out_size, void* d_ws, size_t ws_size,
                              hipStream_t stream) {
  (void)in_sizes; (void)n_in; (void)out_size; (void)ws_size;
  const float* coords = (const float*)d_in[0];
  // d_in[1] = num_atoms (always NATOMS) — unused

  float* out  = (float*)d_out;
  float* H    = out;                                  // B*TN*TN
  float* disp = out + (size_t)BN * TN * TN;           // B*TN
  float* vol  = disp + (size_t)BN * TN;               // B*3*BOX3
  float* lam  = vol + (size_t)BN * 3 * BOX3;          // B

  float* X = (float*)d_ws;                            // B*TN*NV
  float* Y = X + (size_t)BN * TN * NV;                // B*TN*NV

  hess_kernel<<<dim3((BN * NATOMS + 3) / 4), 128, 0, stream>>>(coords, H);
  init_x_kernel<<<(BN * TN * NV + 255) / 256, 256, 0, stream>>>(X);
  ortho_kernel<<<BN, 256, 0, stream>>>(X, X);
  for (int it = 0; it < NITER; ++it) {
    gemm_kernel<<<dim3(TN / (16 * WPB), BN), WPB * 32, 0, stream>>>(H, X, Y);
    ortho_kernel<<<BN, 256, 0, stream>>>(Y, X);
  }
  finalize_kernel<<<BN, 256, 0, stream>>>(H, X, disp, lam);
  zero_vol_kernel<<<(BN * 3 * BOX3 + 255) / 256, 256, 0, stream>>>(vol);
  raster_kernel<<<(BN * NATOMS + 255) / 256, 256, 0, stream>>>(coords, disp, vol);
}